// SimpleMamba_61950608277755
// MI455X (gfx1250) — compile-verified
//
#include <hip/hip_runtime.h>

// MI455X / gfx1250 implementation of the SimpleMamba head.
//
// Structure: only out[:, -1, :] feeds the final prediction, so the per-step
// C-contraction and out/fc projections collapse into a per-batch epilogue.
// The two real GEMMs (in_proj: [B*L,12]x[12,48], x_proj: [B*L,24]x[24,36])
// are done with V_WMMA_F32_16X16X4_F32 (one 16x16 C tile per wave32, K
// stepped by 4).  The sequential SSM scan runs with one block per batch, one
// thread per (d_inner, d_state) state element; the time loop is unrolled so
// the only true dependency chain is one FMA per step (softplus/exp/loads of
// future steps are h-independent and get scheduled ahead of the chain).

typedef float v2f __attribute__((ext_vector_type(2)));
typedef float v8f __attribute__((ext_vector_type(8)));

constexpr int kB      = 64;
constexpr int kSeq    = 2048;
constexpr int kDm     = 12;      // d_model
constexpr int kDi     = 24;      // d_inner
constexpr int kDs     = 16;      // d_state
constexpr int kDc     = 4;       // d_conv
constexpr int kDtr    = 4;       // dt_rank
constexpr int kRows   = kB * kSeq;         // 131072 GEMM rows
constexpr int kXdblC  = kDtr + 2 * kDs;    // 36 columns of x_dbl

__device__ __forceinline__ float sigmoidf_(float v) {
    return 1.0f / (1.0f + __expf(-v));
}

// ---------------------------------------------------------------------------
// K1: xz = x @ in_proj_w^T  via V_WMMA_F32_16X16X4_F32.
//     Store cols 0..23 (x_in) for all rows; cols 24..47 (z) only at t==L-1.
// One wave handles one (Mtile 16 rows) x (Ntile 16 cols) tile, K=12 (3 steps).
// ---------------------------------------------------------------------------
__global__ void k_inproj(const float* __restrict__ x,     // [kRows, 12]
                         const float* __restrict__ Wi,    // [48, 12]
                         float* __restrict__ xin,         // [kRows, 24]
                         float* __restrict__ zlast)       // [64, 24]
{
    const int lane    = threadIdx.x & 31;
    const int wave    = threadIdx.x >> 5;
    const int task    = blockIdx.x * 8 + wave;     // 8 waves per block
    const int tileN   = task % 3;                  // 3 N-tiles cover 48 cols
    const int tileM   = task / 3;
    const int rowbase = tileM * 16;
    const int n0      = tileN * 16;
    const int half    = lane >> 4;                 // 0: K+0/1, 1: K+2/3
    const int l15     = lane & 15;
    const int arow    = rowbase + l15;             // A: M = lane&15 both halves
    const int ncol    = n0 + l15;                  // B: N = lane&15 both halves

    v8f c = {};
    #pragma unroll
    for (int kk = 0; kk < kDm; kk += 4) {
        const int kb = kk + (half << 1);
        v2f a, bm;
        a[0]  = x[arow * kDm + kb];
        a[1]  = x[arow * kDm + kb + 1];
        bm[0] = Wi[ncol * kDm + kb];               // B[k][n] = Wi[n][k]
        bm[1] = Wi[ncol * kDm + kb + 1];
        c = __builtin_amdgcn_wmma_f32_16x16x4_f32(false, a, false, bm,
                                                  (short)0, c, false, false);
    }
    // C/D layout: VGPR r -> M = r (lanes 0-15) / r+8 (lanes 16-31), N = lane&15
    #pragma unroll
    for (int r = 0; r < 8; ++r) {
        const int row = rowbase + r + (half << 3);
        const int col = n0 + l15;
        const float v = c[r];
        if (col < kDi) {
            xin[row * kDi + col] = v;
        } else if ((row & (kSeq - 1)) == kSeq - 1) {
            zlast[(row >> 11) * kDi + (col - kDi)] = v;   // row>>11 == batch
        }
    }
}

// ---------------------------------------------------------------------------
// K2: causal depthwise conv1d (width 4) + bias + SiLU  ->  xc
// ---------------------------------------------------------------------------
__global__ void k_conv(const float* __restrict__ xin,     // [kRows, 24]
                       const float* __restrict__ cw,      // [24, 1, 4]
                       const float* __restrict__ cb,      // [24]
                       float* __restrict__ xc)            // [kRows, 24]
{
    const int idx = blockIdx.x * 256 + threadIdx.x;       // kRows*24 threads
    const int d   = idx % kDi;
    const int row = idx / kDi;
    const int t   = row & (kSeq - 1);
    float acc = cb[d];
    #pragma unroll
    for (int j = 0; j < kDc; ++j) {
        const int tj = t - (kDc - 1) + j;
        if (tj >= 0)
            acc += cw[d * kDc + j] * xin[(row - (kDc - 1) + j) * kDi + d];
    }
    xc[idx] = acc * sigmoidf_(acc);                       // SiLU
}

// ---------------------------------------------------------------------------
// K3: x_dbl = xc @ x_proj_w^T  via V_WMMA_F32_16X16X4_F32.  K=24 (6 steps),
//     N=36 padded to 48 (3 tiles), OOB weight columns read as 0 / not stored.
// ---------------------------------------------------------------------------
__global__ void k_xproj(const float* __restrict__ xc,     // [kRows, 24]
                        const float* __restrict__ Wx,     // [36, 24]
                        float* __restrict__ xdbl)         // [kRows, 36]
{
    const int lane    = threadIdx.x & 31;
    const int wave    = threadIdx.x >> 5;
    const int task    = blockIdx.x * 8 + wave;
    const int tileN   = task % 3;
    const int tileM   = task / 3;
    const int rowbase = tileM * 16;
    const int n0      = tileN * 16;
    const int half    = lane >> 4;
    const int l15     = lane & 15;
    const int arow    = rowbase + l15;
    const int ncol    = n0 + l15;
    const bool nok    = (ncol < kXdblC);

    v8f c = {};
    #pragma unroll
    for (int kk = 0; kk < kDi; kk += 4) {
        const int kb = kk + (half << 1);
        v2f a, bm;
        a[0]  = xc[arow * kDi + kb];
        a[1]  = xc[arow * kDi + kb + 1];
        bm[0] = nok ? Wx[ncol * kDi + kb]     : 0.0f;
        bm[1] = nok ? Wx[ncol * kDi + kb + 1] : 0.0f;
        c = __builtin_amdgcn_wmma_f32_16x16x4_f32(false, a, false, bm,
                                                  (short)0, c, false, false);
    }
    #pragma unroll
    for (int r = 0; r < 8; ++r) {
        const int row = rowbase + r + (half << 3);
        const int col = n0 + l15;
        if (col < kXdblC) xdbl[row * kXdblC + col] = c[r];
    }
}

// ---------------------------------------------------------------------------
// K4: sequential SSM scan.  One block per batch, thread tid = d*16 + s owns
//     state h[d][s].  Per step: dt = softplus(dt_raw @ dt_proj_w^T + b) (4
//     FMAs, duplicated across the 16 s-threads of a channel — cheaper than a
//     barrier), then h = exp(dt*A)*h + dt*B[s]*xc[d].  The time loop is
//     unrolled x4 so all h-independent work (loads, softplus, exp) pipelines
//     ahead of the single chained FMA per step.  Epilogue contracts with C at
//     t=L-1, applies +xc*D, *silu(z), and the folded (fc_w@out_proj_w) g[d].
// ---------------------------------------------------------------------------
__global__ void k_scan(const float* __restrict__ xc,      // [kRows, 24]
                       const float* __restrict__ xdbl,    // [kRows, 36]
                       const float* __restrict__ zlast,   // [64, 24]
                       const float* __restrict__ dtw,     // [24, 4]
                       const float* __restrict__ dtb,     // [24]
                       const float* __restrict__ Alog,    // [24, 16]
                       const float* __restrict__ Dp,      // [24]
                       const float* __restrict__ Wo,      // [12, 24]
                       const float* __restrict__ fcw,     // [1, 12]
                       const float* __restrict__ fcb,     // [1]
                       float* __restrict__ out)           // [64]
{
    const int  b   = blockIdx.x;
    const int  tid = threadIdx.x;          // 384 = 24*16
    const int  d   = tid >> 4;
    const int  s   = tid & 15;
    const float A_ds = -__expf(Alog[d * kDs + s]);
    const float w0 = dtw[d * kDtr + 0], w1 = dtw[d * kDtr + 1];
    const float w2 = dtw[d * kDtr + 2], w3 = dtw[d * kDtr + 3];
    const float bb = dtb[d];
    const long bbase = (long)b * kSeq;

    float h = 0.0f;
    #pragma unroll 4
    for (int t = 0; t < kSeq; ++t) {
        const long base = bbase + t;
        // x_dbl row is 36 floats (144B) -> first 4 are 16B-aligned
        const float4 r = *reinterpret_cast<const float4*>(xdbl + base * kXdblC);
        const float pre = r.x * w0 + r.y * w1 + r.z * w2 + r.w * w3 + bb;
        const float dtv = (pre > 20.0f) ? pre : log1pf(__expf(pre));
        const float Bv  = xdbl[base * kXdblC + kDtr + s];
        const float xv  = xc[base * kDi + d];
        const float dec = __expf(dtv * A_ds);     // off the recurrence chain
        const float add = (dtv * Bv) * xv;        // off the recurrence chain
        h = fmaf(dec, h, add);                    // only chained op per step
    }

    __shared__ float red[kDi * kDs];
    __shared__ float red2[kDi];
    const float Cs = xdbl[(bbase + kSeq - 1) * kXdblC + kDtr + kDs + s];
    red[tid] = h * Cs;
    __syncthreads();

    if (tid < kDi) {
        float y = 0.0f;
        #pragma unroll
        for (int ss = 0; ss < kDs; ++ss) y += red[tid * kDs + ss];
        y += xc[(bbase + kSeq - 1) * kDi + tid] * Dp[tid];
        const float zv = zlast[b * kDi + tid];
        y *= zv * sigmoidf_(zv);                          // * silu(z)
        float g = 0.0f;                                   // fold fc @ out_proj
        #pragma unroll
        for (int e = 0; e < kDm; ++e) g += fcw[e] * Wo[e * kDi + tid];
        red2[tid] = y * g;
    }
    __syncthreads();
    if (tid == 0) {
        float acc = fcb[0];
        #pragma unroll
        for (int dd = 0; dd < kDi; ++dd) acc += red2[dd];
        out[b] = acc;
    }
}

// ---------------------------------------------------------------------------
extern "C" void kernel_launch(void* const* d_in, const int* in_sizes, int n_in,
                              void* d_out, int out_size, void* d_ws, size_t ws_size,
                              hipStream_t stream) {
    (void)in_sizes; (void)n_in; (void)out_size; (void)ws_size;
    const float* x    = (const float*)d_in[0];
    const float* Wi   = (const float*)d_in[1];
    const float* cw   = (const float*)d_in[2];
    const float* cb   = (const float*)d_in[3];
    const float* Wx   = (const float*)d_in[4];
    const float* dtw  = (const float*)d_in[5];
    const float* dtbv = (const float*)d_in[6];
    const float* Alog = (const float*)d_in[7];
    const float* Dp   = (const float*)d_in[8];
    const float* Wo   = (const float*)d_in[9];
    const float* fcw  = (const float*)d_in[10];
    const float* fcb  = (const float*)d_in[11];
    float* out = (float*)d_out;

    // Workspace layout (floats); ~42 MB total, all fully written before read.
    float* ws    = (float*)d_ws;
    float* xc    = ws;                                   // kRows*24
    float* xin   = ws + (size_t)kRows * kDi;             // kRows*24
    float* xdbl  = ws + (size_t)2 * kRows * kDi;         // kRows*36
    float* zlast = xdbl + (size_t)kRows * kXdblC;        // 64*24

    const int gemmBlocks = (kRows / 16) * 3 / 8;         // 3072 blocks, 8 waves each
    k_inproj<<<gemmBlocks, 256, 0, stream>>>(x, Wi, xin, zlast);
    k_conv  <<<(kRows * kDi) / 256, 256, 0, stream>>>(xin, cw, cb, xc);
    k_xproj <<<gemmBlocks, 256, 0, stream>>>(xc, Wx, xdbl);
    k_scan  <<<kB, kDi * kDs, 0, stream>>>(xc, xdbl, zlast, dtw, dtbv, Alog,
                                           Dp, Wo, fcw, fcb, out);
}